// JointAngleLoss_20418274525538
// MI455X (gfx1250) — compile-verified
//
#include <hip/hip_runtime.h>
#include <hip/hip_bf16.h>

#define WEIGHT_C 0.02f
#define EPS_C    1e-6f

typedef __attribute__((ext_vector_type(2))) float v2f;
typedef __attribute__((ext_vector_type(8))) float v8f;

// ---------------------------------------------------------------------------
// Kernel 1: one block per batch element. Each thread computes the angle term
// for one triplet, then a fixed-order LDS tree reduction forms
//   loss[b] = sum(V*w) / max(sum(w), 1)
// Memory-bound: every coord/mask byte is read exactly once.
// ---------------------------------------------------------------------------
__global__ void __launch_bounds__(128) tripletLossPerBatch(
    const float* __restrict__ coords,   // [B, P, 3] f32
    const int*   __restrict__ trip,     // [T, 3]    i32
    const int*   __restrict__ mask,     // [B, T]    i32
    float*       __restrict__ lossOut,  // [B]       f32 (workspace)
    int P, int T) {
  const int b   = blockIdx.x;
  const int tid = threadIdx.x;
  const float* cb = coords + (size_t)b * (size_t)P * 3u;

  float vw = 0.0f, wsum = 0.0f;
  for (int t = tid; t < T; t += (int)blockDim.x) {
    const int i = trip[3 * t + 0];
    const int j = trip[3 * t + 1];
    const int k = trip[3 * t + 2];

    const float pjx = cb[3 * j + 0], pjy = cb[3 * j + 1], pjz = cb[3 * j + 2];
    const float v1x = cb[3 * i + 0] - pjx;
    const float v1y = cb[3 * i + 1] - pjy;
    const float v1z = cb[3 * i + 2] - pjz;
    const float v2x = cb[3 * k + 0] - pjx;
    const float v2y = cb[3 * k + 1] - pjy;
    const float v2z = cb[3 * k + 2] - pjz;

    const float dot = v1x * v2x + v1y * v2y + v1z * v2z;
    const float n1  = sqrtf(v1x * v1x + v1y * v1y + v1z * v1z);
    const float n2  = sqrtf(v2x * v2x + v2y * v2y + v2z * v2z);
    const float V   = fabsf(dot / (n1 * n2 + EPS_C));

    const float w = (float)mask[(size_t)b * (size_t)T + t];
    vw   += V * w;
    wsum += w;
  }

  __shared__ float sVw[128];
  __shared__ float sW[128];
  sVw[tid] = vw;
  sW[tid]  = wsum;
  __syncthreads();
#pragma unroll
  for (int s = 64; s > 0; s >>= 1) {
    if (tid < s) { sVw[tid] += sVw[tid + s]; sW[tid] += sW[tid + s]; }
    __syncthreads();
  }
  if (tid == 0) lossOut[b] = sVw[0] / fmaxf(sW[0], 1.0f);
}

// ---------------------------------------------------------------------------
// Kernel 2: deterministic mean over B values using V_WMMA_F32_16X16X4_F32.
// B = all-ones => D[m][n] = sum_k A[m][k] + C[m][n]; C is the running
// accumulator. A layout (32-bit 16x4): lanes 0-15 hold K=0,1 and lanes 16-31
// hold K=2,3, so each lane loading vals[base+2*lane .. +1] covers all 64
// elements exactly once per step. D layout: lane 0 = rows 0-7 / col 0,
// lane 16 = rows 8-15 / col 0 => wave total = shfl(s,0)+shfl(s,16).
// ---------------------------------------------------------------------------
__global__ void __launch_bounds__(512) reduceMeanWMMA(
    const float* __restrict__ vals, float* __restrict__ out, int B) {
  const int lane = threadIdx.x & 31;
  const int wv   = threadIdx.x >> 5;
  const int nw   = (int)blockDim.x >> 5;

  v8f c = {};
  v2f onesB;
  onesB[0] = 1.0f; onesB[1] = 1.0f;

  for (int base = wv * 64; base < B; base += nw * 64) {
    const int i0 = base + lane * 2;
    v2f a;
    a[0] = (i0     < B) ? vals[i0]     : 0.0f;
    a[1] = (i0 + 1 < B) ? vals[i0 + 1] : 0.0f;
#if __has_builtin(__builtin_amdgcn_wmma_f32_16x16x4_f32)
    c = __builtin_amdgcn_wmma_f32_16x16x4_f32(
        /*neg_a=*/false, a, /*neg_b=*/false, onesB,
        /*c_mod=*/(short)0, c, /*reuse_a=*/false, /*reuse_b=*/false);
#else
    c[0] += a[0] + a[1];   // scalar fallback (still correct, wmma=0 signal)
#endif
  }

  float s = c[0] + c[1] + c[2] + c[3] + c[4] + c[5] + c[6] + c[7];
#if __has_builtin(__builtin_amdgcn_wmma_f32_16x16x4_f32)
  const float waveTotal = __shfl(s, 0, 32) + __shfl(s, 16, 32);
#else
  for (int off = 16; off > 0; off >>= 1) s += __shfl_down(s, off, 32);
  const float waveTotal = __shfl(s, 0, 32);
#endif

  __shared__ float part[32];
  if (lane == 0) part[wv] = waveTotal;
  __syncthreads();
  if (threadIdx.x == 0) {
    float tot = 0.0f;
    for (int w = 0; w < nw; ++w) tot += part[w];   // fixed order: deterministic
    out[0] = WEIGHT_C * (tot / (float)B);
  }
}

// ---------------------------------------------------------------------------
extern "C" void kernel_launch(void* const* d_in, const int* in_sizes, int n_in,
                              void* d_out, int out_size, void* d_ws, size_t ws_size,
                              hipStream_t stream) {
  const float* coords = (const float*)d_in[0];   // [B, P, 3] f32
  const int*   trip   = (const int*)d_in[1];     // [T, 3]    i32
  const int*   mask   = (const int*)d_in[2];     // [B, T]    i32
  float*       out    = (float*)d_out;

  const int T = in_sizes[1] / 3;                 // 100
  const int B = in_sizes[2] / T;                 // 32768
  const int P = in_sizes[0] / (B * 3);           // 308

  float* lossWs = (float*)d_ws;                  // B floats of scratch

  tripletLossPerBatch<<<B, 128, 0, stream>>>(coords, trip, mask, lossWs, P, T);
  reduceMeanWMMA<<<1, 512, 0, stream>>>(lossWs, out, B);
}